// SSIMLoss_44616120271363
// MI455X (gfx1250) — compile-verified
//
#include <hip/hip_runtime.h>

typedef float v2f __attribute__((ext_vector_type(2)));
typedef float v8f __attribute__((ext_vector_type(8)));

#define IMG_N     64
#define IMG_HW    512
#define WAVES     4
#define THREADS   (WAVES * 32)
#define N_TILES   (IMG_N * 32 * 32)        // 65536 16x16 tiles
#define N_BLOCKS  (N_TILES / WAVES)        // 16384
#define AT_STRIDE 28                       // 16 + 2*5 halo, padded to K-chunk 28
#define AT_ROWS   32                       // 26 rows used, padded for row-group 1

__device__ __forceinline__ float gval(const float* gs, int i) {
  return ((unsigned)i < 11u) ? gs[i] : 0.0f;
}

__device__ __forceinline__ v8f wmma4(v2f a, v2f b, v8f c) {
  // D = A(16x4,f32) * B(4x16,f32) + C(16x16,f32)
  return __builtin_amdgcn_wmma_f32_16x16x4_f32(false, a, false, b,
                                               (short)0, c, false, false);
}

// One separable 11-tap Gaussian conv (horizontal WMMA pass -> LDS -> vertical
// WMMA pass) for one of the 5 SSIM maps, for this wave's 16x16 tile.
template <int MAP>
__device__ __forceinline__ v8f conv_map(const float* at, const float* bt,
                                        float* Tw, const v2f* Gf, int lane) {
  const int nlo = lane & 15;
  const int kk  = (lane >> 4) << 1;
  const int r0  = nlo;
  const int r1  = 16 + nlo;

  v8f h0 = {};
  v8f h1 = {};
#pragma unroll
  for (int c = 0; c < 7; ++c) {
    const int col = 4 * c + kk;
    v2f p0, p1;
    if (MAP == 0) {
      p0 = *(const v2f*)(at + r0 * AT_STRIDE + col);
      p1 = *(const v2f*)(at + r1 * AT_STRIDE + col);
    } else if (MAP == 1) {
      p0 = *(const v2f*)(bt + r0 * AT_STRIDE + col);
      p1 = *(const v2f*)(bt + r1 * AT_STRIDE + col);
    } else if (MAP == 2) {
      v2f a0 = *(const v2f*)(at + r0 * AT_STRIDE + col);
      v2f a1 = *(const v2f*)(at + r1 * AT_STRIDE + col);
      p0 = a0 * a0; p1 = a1 * a1;
    } else if (MAP == 3) {
      v2f b0 = *(const v2f*)(bt + r0 * AT_STRIDE + col);
      v2f b1 = *(const v2f*)(bt + r1 * AT_STRIDE + col);
      p0 = b0 * b0; p1 = b1 * b1;
    } else {
      v2f a0 = *(const v2f*)(at + r0 * AT_STRIDE + col);
      v2f a1 = *(const v2f*)(at + r1 * AT_STRIDE + col);
      v2f b0 = *(const v2f*)(bt + r0 * AT_STRIDE + col);
      v2f b1 = *(const v2f*)(bt + r1 * AT_STRIDE + col);
      p0 = a0 * b0; p1 = a1 * b1;
    }
    h0 = wmma4(p0, Gf[c], h0);
    h1 = wmma4(p1, Gf[c], h1);
  }

  // Scatter D fragments (rows 0..31 of H) into LDS scratch.
  const int rofs = (lane >= 16) ? 8 : 0;
#pragma unroll
  for (int j = 0; j < 8; ++j) {
    Tw[(j + rofs) * 16 + nlo]      = h0[j];
    Tw[(16 + j + rofs) * 16 + nlo] = h1[j];
  }
  __syncthreads();

  // Vertical pass: D = G(16x26) * H(26x16); rows >=26 hit exact zeros in Gf.
  v8f acc = {};
#pragma unroll
  for (int c = 0; c < 7; ++c) {
    const int k = 4 * c + kk;
    v2f bb;
    bb.x = Tw[k * 16 + nlo];
    bb.y = Tw[(k + 1) * 16 + nlo];
    acc = wmma4(Gf[c], bb, acc);
  }
  __syncthreads();  // Tw reused by next map
  return acc;
}

__global__ void __launch_bounds__(THREADS)
ssim_tiles(const float* __restrict__ x_hat, const float* __restrict__ x,
           const float* __restrict__ kern, float* __restrict__ part) {
  __shared__ float g_s[16];
  __shared__ float a_sh[WAVES][AT_ROWS * AT_STRIDE];
  __shared__ float b_sh[WAVES][AT_ROWS * AT_STRIDE];
  __shared__ float t_sh[WAVES][32 * 16];
  __shared__ float wsum[WAVES];

  const int tid  = threadIdx.x;
  const int w    = tid >> 5;
  const int lane = tid & 31;

  // 1D Gaussian = row sums of the normalized 2D kernel (rank-1, sum==1).
  if (tid < 16) {
    float s = 0.0f;
    if (tid < 11) {
#pragma unroll
      for (int j = 0; j < 11; ++j) s += kern[tid * 11 + j];
    }
    g_s[tid] = s;
  }
  __syncthreads();

  const int tile = blockIdx.x * WAVES + w;
  const int img  = tile >> 10;
  const int rem  = tile & 1023;
  const int y0   = (rem >> 5) << 4;
  const int x0   = (rem & 31) << 4;

  float* at = a_sh[w];
  float* bt = b_sh[w];
  float* Tw = t_sh[w];

  const float* xp  = x     + (size_t)img * (IMG_HW * IMG_HW);
  const float* xhp = x_hat + (size_t)img * (IMG_HW * IMG_HW);

  // Load haloed tile, apply [-1,1] -> [0,1] shift, zero-pad borders.
  for (int i = lane; i < AT_ROWS * AT_STRIDE; i += 32) {
    const int r  = i / AT_STRIDE;
    const int c  = i % AT_STRIDE;
    const int gy = y0 + r - 5;
    const int gx = x0 + c - 5;
    float va = 0.0f, vb = 0.0f;
    if (gy >= 0 && gy < IMG_HW && gx >= 0 && gx < IMG_HW) {
      const int off = gy * IMG_HW + gx;
      va = 0.5f * (xp[off]  + 1.0f);   // 'x' in _ssim
      vb = 0.5f * (xhp[off] + 1.0f);   // 'y' in _ssim
    }
    at[i] = va;
    bt[i] = vb;
  }
  __syncthreads();

  // Banded Gaussian fragments; identical for A-of-vertical and B-of-horizontal:
  // frag[c] lane L -> g[4c + 2*(L>=16) - (L&15)], g[4c + 2*(L>=16)+1 - (L&15)]
  const int nlo = lane & 15;
  const int kk  = (lane >> 4) << 1;
  v2f Gf[7];
#pragma unroll
  for (int c = 0; c < 7; ++c) {
    const int k = 4 * c + kk;
    v2f t;
    t.x = gval(g_s, k - nlo);
    t.y = gval(g_s, k + 1 - nlo);
    Gf[c] = t;
  }

  const v8f rA  = conv_map<0>(at, bt, Tw, Gf, lane);  // mu_x
  const v8f rB  = conv_map<1>(at, bt, Tw, Gf, lane);  // mu_y
  const v8f rAA = conv_map<2>(at, bt, Tw, Gf, lane);  // E[x^2]
  const v8f rBB = conv_map<3>(at, bt, Tw, Gf, lane);  // E[y^2]
  const v8f rAB = conv_map<4>(at, bt, Tw, Gf, lane);  // E[xy]

  const float C1 = 1e-4f;   // 0.01^2
  const float C2 = 9e-4f;   // 0.03^2
  float lsum = 0.0f;
#pragma unroll
  for (int e = 0; e < 8; ++e) {
    const float mx  = rA[e], my = rB[e];
    const float mx2 = mx * mx, my2 = my * my, mxy = mx * my;
    const float sx  = rAA[e] - mx2;
    const float sy  = rBB[e] - my2;
    const float sxy = rAB[e] - mxy;
    const float num = (2.0f * mxy + C1) * (2.0f * sxy + C2);
    const float den = (mx2 + my2 + C1) * (sx + sy + C2);
    lsum += num / (den + 1e-8f);
  }

  // wave32 reduction, then per-block partial (fully deterministic).
#pragma unroll
  for (int off = 16; off > 0; off >>= 1) lsum += __shfl_down(lsum, off, 32);
  if (lane == 0) wsum[w] = lsum;
  __syncthreads();
  if (tid == 0) {
    float s = 0.0f;
#pragma unroll
    for (int i = 0; i < WAVES; ++i) s += wsum[i];
    part[blockIdx.x] = s;
  }
}

__global__ void __launch_bounds__(256)
ssim_reduce(const float* __restrict__ part, float* __restrict__ out,
            int nparts, float invN) {
  __shared__ float sh[256];
  float s = 0.0f;
  for (int i = threadIdx.x; i < nparts; i += 256) s += part[i];
  sh[threadIdx.x] = s;
  __syncthreads();
  for (int st = 128; st > 0; st >>= 1) {
    if (threadIdx.x < st) sh[threadIdx.x] += sh[threadIdx.x + st];
    __syncthreads();
  }
  if (threadIdx.x == 0) out[0] = 1.0f - sh[0] * invN;
}

extern "C" void kernel_launch(void* const* d_in, const int* in_sizes, int n_in,
                              void* d_out, int out_size, void* d_ws,
                              size_t ws_size, hipStream_t stream) {
  const float* x_hat = (const float*)d_in[0];
  const float* x     = (const float*)d_in[1];
  const float* kern  = (const float*)d_in[2];
  float* part = (float*)d_ws;  // N_BLOCKS floats = 64 KB scratch

  ssim_tiles<<<N_BLOCKS, THREADS, 0, stream>>>(x_hat, x, kern, part);

  const float invN = 1.0f / (float)(IMG_N * IMG_HW * IMG_HW);
  ssim_reduce<<<1, 256, 0, stream>>>(part, (float*)d_out, N_BLOCKS, invN);
}